// G_Nbody_43379169689789
// MI455X (gfx1250) — compile-verified
//
#include <hip/hip_runtime.h>

// ---------------------------------------------------------------------------
// N-body symplectic vector field on MI455X (gfx1250), wave32 + WMMA.
//
//   out[b,0:2] = p_b / m[0]
//   out[b,2:4] = sum_a w_ab * m_a*m_b * (q_b - q_a) / ((r+eps)^2 * r)
//   w_ab: 0 (a==b), 2 (|a-b|==1, tril(.,1) double-counts neighbors), 1 else
//
// 256 blocks x 256 threads. Block owns a 16-body column tile; its 8 wave32s
// split the 256 row tiles (32 each, counted loop -> scalar control flow).
// All 4096 bodies staged to LDS (64KB) via async global->LDS b128 copies;
// masses folded into the dead .w lane. Per row tile one
// V_WMMA_F32_16X16X4_F32 builds the 16x16 squared-distance tile on the
// matrix pipe; the branchless sqrt/rcp force kernel runs on VALU/TRANS over
// the 8 D-layout elements (independent chains -> dual-issue), and column
// forces reduce lane-locally + shfl_xor(16) + LDS across waves.
//
// Diagonal safety is arithmetic (denominator floored at 1e-30 so rcp stays
// finite; weight 0 annihilates it exactly) -- no divergent branches, EXEC
// stays all-ones through the whole WMMA loop.
// ---------------------------------------------------------------------------

typedef float v2f __attribute__((ext_vector_type(2)));
typedef float v8f __attribute__((ext_vector_type(8)));

#define N_BODIES 4096
#define TILE     16
#define N_TILES  (N_BODIES / TILE)   // 256
#define THREADS  256
#define WAVES    (THREADS / 32)      // 8 waves (wave32)
#define TPW      (N_TILES / WAVES)   // 32 row tiles per wave
#define GEPS     1e-10f
#define DEN_FLOOR 1e-30f

__global__ __launch_bounds__(THREADS)
void nbody_symp_kernel(const float4* __restrict__ xg,  // [N] (qx,qy,px,py)
                       const float*  __restrict__ mg,  // [N]
                       float4*       __restrict__ outg)
{
    __shared__ float4 sx[N_BODIES];  // staged as (qx, qy, px(dead), m) : 64KB

    const int tid  = threadIdx.x;
    const int lane = tid & 31;
    const int wv   = tid >> 5;
    const int h    = lane >> 4;      // wave half (WMMA layout)
    const int ln   = lane & 15;

    // ---- stage x[] into LDS with gfx1250 async global->LDS copies ----------
    const float* xf = (const float*)xg;
    float mv[16];
#pragma unroll
    for (int k = 0; k < 16; ++k) mv[k] = mg[tid + 256 * k];   // overlap w/ async
#pragma unroll
    for (int k = 0; k < 16; ++k) {
        const int i = tid + 256 * k;
        unsigned ldsoff = (unsigned)(unsigned long long)(&sx[i]); // LDS byte off
        unsigned voff   = (unsigned)(i * 16);
        asm volatile("global_load_async_to_lds_b128 %0, %1, %2"
                     :: "v"(ldsoff), "v"(voff), "s"(xf) : "memory");
    }
    asm volatile("s_wait_asynccnt 0x0" ::: "memory");
    __syncthreads();
    // fold node masses into the unused .w slot (px in .z stays dead)
#pragma unroll
    for (int k = 0; k < 16; ++k) {
        const int i = tid + 256 * k;
        ((float*)&sx[i])[3] = mv[k];
    }
    __syncthreads();

    // ---- this wave's output column bodies ----------------------------------
    const int    bglob = blockIdx.x * TILE + ln;     // lanes l and l^16 share b
    const float4 xb    = xg[bglob];
    const float  mb    = mg[bglob];
    const float  qxb = xb.x, qyb = xb.y;
    const float  n2b = qxb * qxb + qyb * qyb;

    // B (4x16): rows (-2qx_b, -2qy_b, 1, n2_b); K striped across halves
    v2f Bm;
    Bm[0] = h ? 1.0f : (-2.0f * qxb);
    Bm[1] = h ? n2b  : (-2.0f * qyb);

    float fx = 0.0f, fy = 0.0f;

    for (int t = 0; t < TPW; ++t) {                  // counted: scalar control
        const int jt    = wv + t * WAVES;
        const int rbase = jt * TILE;

        // A (16x4): row a = (qx_a, qy_a, n2_a, 1); lane m -> K0,K1; m+16 -> K2,K3
        const float2 qa2 = *(const float2*)(&sx[rbase + ln]);
        const float  n2a = qa2.x * qa2.x + qa2.y * qa2.y;
        v2f Am;
        Am[0] = h ? n2a  : qa2.x;
        Am[1] = h ? 1.0f : qa2.y;

        v8f acc = {};
        // D[a,b] = -2 qa.qb + |qa|^2 + |qb|^2 = r^2  (16x16 tile, one WMMA)
        acc = __builtin_amdgcn_wmma_f32_16x16x4_f32(
                  false, Am, false, Bm, (short)0, acc, false, false);

#pragma unroll
        for (int v = 0; v < 8; ++v) {
            const int    arow = rbase + 8 * h + v;   // D layout: M = v + 8h
            const float4 xa   = sx[arow];            // broadcast within half
            const float  dx   = qxb - xa.x;
            const float  dy   = qyb - xa.y;
            const float  r2   = fmaxf(acc[v], 0.0f); // guard Gram cancellation
            const float  r    = __builtin_amdgcn_sqrtf(r2);
            const float  tt   = r + GEPS;
            // floored denominator: rcp stays finite on the diagonal (r==0),
            // and w==0 annihilates it exactly -> branchless, no NaN.
            const float  den  = fmaf(tt * tt, r, DEN_FLOOR);
            const float  base = mb * xa.w * __builtin_amdgcn_rcpf(den);
            const int    dlt  = arow - bglob;
            const int    ad   = dlt < 0 ? -dlt : dlt;
            const float  w    = (ad == 1) ? 2.0f : ((ad == 0) ? 0.0f : 1.0f);
            const float  g    = w * base;
            fx = fmaf(g, dx, fx);
            fy = fmaf(g, dy, fy);
        }
    }

    // lanes l and l^16 accumulated disjoint row halves of the same column b
    fx += __shfl_xor(fx, 16, 32);
    fy += __shfl_xor(fy, 16, 32);

    __syncthreads();                  // all sx reads done -> reuse LDS
    float2* red = (float2*)sx;
    if (lane < 16) red[wv * 16 + ln] = make_float2(fx, fy);
    __syncthreads();

    if (tid < 16) {
        float FX = 0.0f, FY = 0.0f;
#pragma unroll
        for (int w = 0; w < WAVES; ++w) {
            const float2 pr = red[w * 16 + tid];
            FX += pr.x;
            FY += pr.y;
        }
        const int    body  = blockIdx.x * TILE + tid;
        const float4 xb2   = xg[body];
        const float  invm0 = 1.0f / mg[0];
        float4 o;
        o.x = xb2.z * invm0;   // dH/dp = p / m[0]
        o.y = xb2.w * invm0;
        o.z = FX;              // -dH/dq
        o.w = FY;
        outg[body] = o;
    }
}

extern "C" void kernel_launch(void* const* d_in, const int* in_sizes, int n_in,
                              void* d_out, int out_size, void* d_ws, size_t ws_size,
                              hipStream_t stream) {
    (void)in_sizes; (void)n_in; (void)out_size; (void)d_ws; (void)ws_size;
    const float4* x = (const float4*)d_in[0];
    const float*  m = (const float*)d_in[1];
    nbody_symp_kernel<<<N_TILES, THREADS, 0, stream>>>(x, m, (float4*)d_out);
}